// DualTowerTitans_10840497455734
// MI455X (gfx1250) — compile-verified
//
#include <hip/hip_runtime.h>
#include <math.h>

// ---------------------------------------------------------------------------
// DualTowerTitans forward for MI455X (gfx1250, wave32).
//
// Roofline: state tensor traffic dominates (2 reads + 1 write ~= 805MB ->
// ~35us at 23.3TB/s). GEMM FLOPs (~3.5GF) are noise and L2-resident.
//  - dense GEMMs: V_WMMA_F32_16X16X4_F32, one wave per 16x16 tile,
//    branch-free inner loop (clamped B-row, stores guarded in epilogue).
//  - state passes: per-wave TDM (TENSOR_LOAD_TO_LDS / TENSOR_STORE_FROM_LDS)
//    of 32-row slices, compute from LDS, single-pass update (butterfly
//    reduction broadcasts pred to all lanes -> no barriers at all).
// ---------------------------------------------------------------------------

#define BX 1024
#define DD 256
#define FFD 2048
#define N2 514              // 2D+2
#define SCALE_ETA 0.0625f   // D^-0.5

typedef __attribute__((ext_vector_type(2))) float v2f;
typedef __attribute__((ext_vector_type(8))) float v8f;
typedef __attribute__((ext_vector_type(4))) unsigned int v4u;
typedef __attribute__((ext_vector_type(8))) int v8i_;
typedef __attribute__((ext_vector_type(4))) int v4i_;

#if __has_builtin(__builtin_amdgcn_tensor_load_to_lds) && \
    __has_builtin(__builtin_amdgcn_tensor_store_from_lds) && \
    __has_builtin(__builtin_amdgcn_s_wait_tensorcnt)
#define USE_TDM 1
#else
#define USE_TDM 0
#endif

__device__ __forceinline__ float wave_sum(float s) {
  #pragma unroll
  for (int off = 16; off > 0; off >>= 1) s += __shfl_xor(s, off, 32);
  return s;
}
__device__ __forceinline__ float sigmoidf_(float x) { return 1.0f / (1.0f + __expf(-x)); }

#if USE_TDM
// ---------------------------------------------------------------------------
// TDM 2D tile copy: `rows` rows of 256 f32 between global and LDS.
// D# packing per CDNA5 ISA:
//   g0: [1:0]=count=1, [63:32]=lds_addr, [120:64]=global_addr, [127:126]=2
//   g1: [17:16]=data_size(2 -> 4B), [79:48]=tensor_dim0, [111:80]=tensor_dim1,
//       [127:112]=tile_dim0, [143:128]=tile_dim1, [207:160]=dim0_stride
// ---------------------------------------------------------------------------
__device__ __forceinline__ void tdm_copy_rows(unsigned int lds_byte,
                                              unsigned long long gbyte,
                                              unsigned int rows, int is_store) {
  v4u g0;
  g0.x = 1u;                                                    // count=1
  g0.y = lds_byte;                                              // lds_addr
  g0.z = (unsigned int)gbyte;                                   // ga[31:0]
  g0.w = (unsigned int)((gbyte >> 32) & 0x01ffffffu) | (2u << 30); // ga[56:32]|type=2
  v8i_ g1;
  g1[0] = 0x00020000;              // data_size = 4 bytes
  g1[1] = (int)(DD << 16);         // tensor_dim0[15:0] @ bits 63:48
  g1[2] = (int)(rows << 16);       // tensor_dim0[31:16]=0 | tensor_dim1[15:0]
  g1[3] = (int)(DD << 16);         // tensor_dim1[31:16]=0 | tile_dim0=256
  g1[4] = (int)rows;               // tile_dim1 ; tile_dim2=0
  g1[5] = (int)DD;                 // tensor_dim0_stride[31:0]
  g1[6] = 0;
  g1[7] = 0;
  v4i_ z = {0, 0, 0, 0};
#if defined(__clang_major__) && (__clang_major__ >= 23)
  v8i_ z8 = {0, 0, 0, 0, 0, 0, 0, 0};
  if (is_store) __builtin_amdgcn_tensor_store_from_lds(g0, g1, z, z, z8, 0);
  else          __builtin_amdgcn_tensor_load_to_lds(g0, g1, z, z, z8, 0);
#else
  if (is_store) __builtin_amdgcn_tensor_store_from_lds(g0, g1, z, z, 0);
  else          __builtin_amdgcn_tensor_load_to_lds(g0, g1, z, z, 0);
#endif
}
#endif  // USE_TDM

// ---------------------------------------------------------------------------
// Generic GEMM: out[M,N] = act( X[M,K] @ W[N,K]^T + bias + residual )
// One wave per 16x16 tile via V_WMMA_F32_16X16X4_F32.  Inner loop is
// branch-free: B-row index clamped to N-1 (always in-bounds), only the
// epilogue stores are guarded (relevant solely for the N=514 GEMM).
// ---------------------------------------------------------------------------
__global__ void dt_gemm_wmma(const float* __restrict__ X, const float* __restrict__ W,
                             const float* __restrict__ bias, const float* __restrict__ residual,
                             float* __restrict__ out, int M, int N, int K, int relu) {
  const int wave = blockIdx.x * (blockDim.x >> 5) + (threadIdx.x >> 5);
  const int lane = threadIdx.x & 31;
  const int ntiles = (N + 15) >> 4;
  const int mtiles = M >> 4;
  if (wave >= mtiles * ntiles) return;                 // wave-uniform exit
  const int mt = wave / ntiles, nt = wave - mt * ntiles;
  const int m0 = mt << 4, n0 = nt << 4;
  const int hi = lane >> 4, lo = lane & 15;

  const int n = n0 + lo;
  const int ncl = (n < N) ? n : (N - 1);               // clamp: loads in-bounds
  const float* arow = X + (size_t)(m0 + lo) * K + 2 * hi;
  const float* brow = W + (size_t)ncl * K + 2 * hi;

  v8f acc = {0.f, 0.f, 0.f, 0.f, 0.f, 0.f, 0.f, 0.f};
#if __has_builtin(__builtin_amdgcn_wmma_f32_16x16x4_f32)
  for (int kk = 0; kk < K; kk += 4) {
    v2f a = *(const v2f*)(arow + kk);
    v2f b = *(const v2f*)(brow + kk);
    acc = __builtin_amdgcn_wmma_f32_16x16x4_f32(false, a, false, b,
                                                (short)0, acc, false, false);
  }
#else
  for (int kk = 0; kk < K; kk += 4) {
    v2f a = *(const v2f*)(arow + kk);
    v2f b = *(const v2f*)(brow + kk);
    acc[0] += a.x * b.x + a.y * b.y;
  }
#endif

  if (n < N) {
    const float bv = bias ? bias[n] : 0.f;
    #pragma unroll
    for (int r = 0; r < 8; ++r) {
      const int m = m0 + r + 8 * hi;
      float v = acc[r] + bv;
      if (residual) v += residual[(size_t)m * N + n];
      if (relu) v = fmaxf(v, 0.f);
      out[(size_t)m * N + n] = v;
    }
  }
}

// ---------------------------------------------------------------------------
// LayerNorm over rows of length 256; one wave per row; optional ReLU.
// ---------------------------------------------------------------------------
__global__ void dt_layernorm(const float* __restrict__ in, const float* __restrict__ g,
                             const float* __restrict__ bv, float* __restrict__ out,
                             int rows, int relu) {
  const int row = blockIdx.x * (blockDim.x >> 5) + (threadIdx.x >> 5);
  const int lane = threadIdx.x & 31;
  if (row >= rows) return;
  const float* p = in + (size_t)row * DD + lane * 8;
  float vals[8];
  float s = 0.f, ss = 0.f;
  #pragma unroll
  for (int j = 0; j < 8; ++j) { vals[j] = p[j]; s += vals[j]; ss += vals[j] * vals[j]; }
  s = wave_sum(s); ss = wave_sum(ss);
  const float mean = s * (1.0f / DD);
  const float var  = ss * (1.0f / DD) - mean * mean;
  const float inv  = rsqrtf(var + 1e-5f);
  float* o = out + (size_t)row * DD + lane * 8;
  #pragma unroll
  for (int j = 0; j < 8; ++j) {
    const int c = lane * 8 + j;
    float v = (vals[j] - mean) * inv * g[c] + bv[c];
    if (relu) v = fmaxf(v, 0.f);
    o[j] = v;
  }
}

// Row-wise L2 normalize in place (rows of 256); one wave per row.
__global__ void dt_l2norm(float* __restrict__ buf, int rows) {
  const int row = blockIdx.x * (blockDim.x >> 5) + (threadIdx.x >> 5);
  const int lane = threadIdx.x & 31;
  if (row >= rows) return;
  float* p = buf + (size_t)row * DD + lane * 8;
  float vals[8];
  float ss = 0.f;
  #pragma unroll
  for (int j = 0; j < 8; ++j) { vals[j] = p[j]; ss += vals[j] * vals[j]; }
  ss = wave_sum(ss);
  const float d = 1.0f / fmaxf(sqrtf(ss), 1e-12f);
  #pragma unroll
  for (int j = 0; j < 8; ++j) p[j] = vals[j] * d;
}

// ---------------------------------------------------------------------------
// Pass 1: read[b,i] = sum_j state[b,i,j]*q[b,j].
// One block per b; each wave TDM-loads its 32-row slice (32KB) into LDS,
// waits TENSORcnt, dots out of LDS. q read straight from global (L2-hot).
// No cross-wave sharing -> zero barriers.
// ---------------------------------------------------------------------------
__global__ void dt_state_read(const float* __restrict__ state, const float* __restrict__ q,
                              float* __restrict__ readv) {
  const int b = blockIdx.x;
  const int tid = threadIdx.x;
  const int wave = tid >> 5, lane = tid & 31;
  extern __shared__ float tile[];                 // 256*256 f32 = 256KB
  float qv[8];
  #pragma unroll
  for (int j = 0; j < 8; ++j) qv[j] = q[(size_t)b * DD + lane * 8 + j];

#if USE_TDM
  const unsigned int row0 = (unsigned int)__builtin_amdgcn_readfirstlane(wave) * 32u;
  tdm_copy_rows((unsigned int)(size_t)tile + row0 * DD * 4u,
                (unsigned long long)(size_t)(state + (size_t)b * DD * DD + (size_t)row0 * DD),
                32u, 0);
  __builtin_amdgcn_s_wait_tensorcnt(0);
  const float* src = tile;
#else
  const unsigned int row0 = wave * 32u;
  const float* src = state + (size_t)b * DD * DD;   // direct-global fallback
#endif

  for (int i = (int)row0; i < (int)row0 + 32; ++i) {
    const float* rp = src + (size_t)i * DD + lane * 8;
    float4 a = *(const float4*)(rp);
    float4 c = *(const float4*)(rp + 4);
    float s = a.x * qv[0] + a.y * qv[1] + a.z * qv[2] + a.w * qv[3]
            + c.x * qv[4] + c.y * qv[5] + c.z * qv[6] + c.w * qv[7];
    s = wave_sum(s);
    if (lane == 0) readv[(size_t)b * DD + i] = s;
  }
}

// ---------------------------------------------------------------------------
// Pass 2, single-pass: per wave slice of 32 rows,
//   TDM load -> for each row: pred = dot(row,k) (butterfly leaves the sum in
//   ALL lanes) -> err = v - pred -> row = (1-a)*row + eta*err*k, rewritten in
//   LDS -> TDM store slice to new_state.  No barriers, state 1R+1W in HBM.
// ---------------------------------------------------------------------------
__global__ void dt_state_update(const float* __restrict__ state, const float* __restrict__ kv,
                                const float* __restrict__ vv, const float* __restrict__ alpha,
                                const float* __restrict__ eta, float* __restrict__ ns) {
  const int b = blockIdx.x;
  const int tid = threadIdx.x;
  const int wave = tid >> 5, lane = tid & 31;
  extern __shared__ float tile[];                 // 256*256 f32 = 256KB
  float kvl[8];
  #pragma unroll
  for (int j = 0; j < 8; ++j) kvl[j] = kv[(size_t)b * DD + lane * 8 + j];
  const float om = 1.0f - alpha[b];
  const float e  = eta[b];

#if USE_TDM
  const unsigned int row0 = (unsigned int)__builtin_amdgcn_readfirstlane(wave) * 32u;
  tdm_copy_rows((unsigned int)(size_t)tile + row0 * DD * 4u,
                (unsigned long long)(size_t)(state + (size_t)b * DD * DD + (size_t)row0 * DD),
                32u, 0);
  __builtin_amdgcn_s_wait_tensorcnt(0);

  for (int i = (int)row0; i < (int)row0 + 32; ++i) {
    float* rp = tile + (size_t)i * DD + lane * 8;
    float4 a = *(const float4*)(rp);
    float4 c = *(const float4*)(rp + 4);
    float s = a.x * kvl[0] + a.y * kvl[1] + a.z * kvl[2] + a.w * kvl[3]
            + c.x * kvl[4] + c.y * kvl[5] + c.z * kvl[6] + c.w * kvl[7];
    s = wave_sum(s);                              // pred_i in every lane
    const float ee = e * (vv[(size_t)b * DD + i] - s);
    float4 oa, oc;
    oa.x = om * a.x + ee * kvl[0];  oa.y = om * a.y + ee * kvl[1];
    oa.z = om * a.z + ee * kvl[2];  oa.w = om * a.w + ee * kvl[3];
    oc.x = om * c.x + ee * kvl[4];  oc.y = om * c.y + ee * kvl[5];
    oc.z = om * c.z + ee * kvl[6];  oc.w = om * c.w + ee * kvl[7];
    *(float4*)(rp)     = oa;
    *(float4*)(rp + 4) = oc;
  }
  tdm_copy_rows((unsigned int)(size_t)tile + row0 * DD * 4u,
                (unsigned long long)(size_t)(ns + (size_t)b * DD * DD + (size_t)row0 * DD),
                32u, 1);
  __builtin_amdgcn_s_wait_tensorcnt(0);
#else
  const unsigned int row0 = wave * 32u;
  const float* sb = state + (size_t)b * DD * DD;
  float* nb = ns + (size_t)b * DD * DD;
  for (int i = (int)row0; i < (int)row0 + 32; ++i) {
    const float* rp = sb + (size_t)i * DD + lane * 8;
    float4 a = *(const float4*)(rp);
    float4 c = *(const float4*)(rp + 4);
    float s = a.x * kvl[0] + a.y * kvl[1] + a.z * kvl[2] + a.w * kvl[3]
            + c.x * kvl[4] + c.y * kvl[5] + c.z * kvl[6] + c.w * kvl[7];
    s = wave_sum(s);
    const float ee = e * (vv[(size_t)b * DD + i] - s);
    float4 oa, oc;
    oa.x = om * a.x + ee * kvl[0];  oa.y = om * a.y + ee * kvl[1];
    oa.z = om * a.z + ee * kvl[2];  oa.w = om * a.w + ee * kvl[3];
    oc.x = om * c.x + ee * kvl[4];  oc.y = om * c.y + ee * kvl[5];
    oc.z = om * c.z + ee * kvl[6];  oc.w = om * c.w + ee * kvl[7];
    float* np = nb + (size_t)i * DD + lane * 8;
    *(float4*)(np)     = oa;
    *(float4*)(np + 4) = oc;
  }
#endif
}

// meta_in = concat(x_norm, read)  (B x 512)
__global__ void dt_concat(const float* __restrict__ a, const float* __restrict__ b,
                          float* __restrict__ out) {
  const size_t idx = (size_t)blockIdx.x * blockDim.x + threadIdx.x;
  const size_t row = idx >> 9, col = idx & 511;
  out[idx] = (col < DD) ? a[row * DD + col] : b[row * DD + (col - DD)];
}

// mod = i_norm * (1 + tanh(gamma)) + beta   with gamma/beta slices of meta_out
__global__ void dt_mod(const float* __restrict__ i_norm, const float* __restrict__ meta_out,
                       float* __restrict__ modv) {
  const size_t idx = (size_t)blockIdx.x * blockDim.x + threadIdx.x;
  const size_t brow = idx >> 8, j = idx & 255;
  const float g  = meta_out[brow * N2 + j];
  const float be = meta_out[brow * N2 + DD + j];
  modv[idx] = i_norm[idx] * (1.0f + tanhf(g)) + be;
}

// alpha/eta: per-b dot(mod, Wa/We) + scalar biases + meta bias cols, sigmoid.
__global__ void dt_alpha_eta(const float* __restrict__ modv, const float* __restrict__ Wa,
                             const float* __restrict__ ba, const float* __restrict__ We,
                             const float* __restrict__ be, const float* __restrict__ meta_out,
                             float* __restrict__ alpha, float* __restrict__ eta) {
  const int b = blockIdx.x * (blockDim.x >> 5) + (threadIdx.x >> 5);
  const int lane = threadIdx.x & 31;
  const float* m = modv + (size_t)b * DD + lane * 8;
  const float* wa = Wa + lane * 8;
  const float* we = We + lane * 8;
  float sa = 0.f, se = 0.f;
  #pragma unroll
  for (int j = 0; j < 8; ++j) { sa += m[j] * wa[j]; se += m[j] * we[j]; }
  sa = wave_sum(sa); se = wave_sum(se);
  if (lane == 0) {
    alpha[b] = sigmoidf_(sa + ba[0] + meta_out[(size_t)b * N2 + 512]);
    eta[b]   = sigmoidf_(se + be[0] + meta_out[(size_t)b * N2 + 513]) * SCALE_ETA;
  }
}

// ---------------------------------------------------------------------------
extern "C" void kernel_launch(void* const* d_in, const int* in_sizes, int n_in,
                              void* d_out, int out_size, void* d_ws, size_t ws_size,
                              hipStream_t stream) {
  (void)in_sizes; (void)n_in; (void)out_size; (void)ws_size;
  const float* u     = (const float*)d_in[0];
  const float* it    = (const float*)d_in[1];
  const float* state = (const float*)d_in[2];
  const float* Wq    = (const float*)d_in[3];
  const float* Wk    = (const float*)d_in[4];
  const float* Wv    = (const float*)d_in[5];
  const float* Wa    = (const float*)d_in[6];
  const float* ba    = (const float*)d_in[7];
  const float* We    = (const float*)d_in[8];
  const float* be    = (const float*)d_in[9];
  const float* Wm1   = (const float*)d_in[10];
  const float* bm1   = (const float*)d_in[11];
  const float* mg    = (const float*)d_in[12];
  const float* mb    = (const float*)d_in[13];
  const float* Wm2   = (const float*)d_in[14];
  const float* bm2   = (const float*)d_in[15];
  const float* Wout  = (const float*)d_in[16];
  const float* bout  = (const float*)d_in[17];
  const float* W1    = (const float*)d_in[18];
  const float* b1    = (const float*)d_in[19];
  const float* W2    = (const float*)d_in[20];
  const float* b2    = (const float*)d_in[21];
  const float* n1g   = (const float*)d_in[22];
  const float* n1b   = (const float*)d_in[23];
  const float* n2g   = (const float*)d_in[24];
  const float* n2b   = (const float*)d_in[25];

  float* out_emb   = (float*)d_out;                     // B*D
  float* out_state = (float*)d_out + (size_t)BX * DD;   // B*D*D

  float* W = (float*)d_ws;
  const size_t BD = (size_t)BX * DD;
  float* x_norm   = W;
  float* i_norm   = W + 1 * BD;
  float* qv       = W + 2 * BD;
  float* readv    = W + 3 * BD;
  float* xbuf     = W + 4 * BD;
  float* x2       = W + 5 * BD;
  float* hbuf     = W + 6 * BD;       // meta hidden (LN+relu in place)
  float* modv     = W + 7 * BD;
  float* kv       = W + 8 * BD;
  float* vv       = W + 9 * BD;
  float* meta_in  = W + 10 * BD;      // B*512
  float* meta_out = W + 12 * BD;      // B*514
  float* alpha    = meta_out + (size_t)BX * N2;
  float* eta      = alpha + BX;
  float* h1       = eta + BX;         // B*FF

  const dim3 blk(256);
  const size_t tile_sh = (size_t)DD * DD * sizeof(float);   // 256KB dynamic LDS
  auto gblk = [](int mtiles, int ntiles) { return dim3((mtiles * ntiles + 7) / 8); };

  dt_layernorm<<<dim3(BX / 8), blk, 0, stream>>>(u,  n1g, n1b, x_norm, BX, 0);
  dt_layernorm<<<dim3(BX / 8), blk, 0, stream>>>(it, n1g, n1b, i_norm, BX, 0);
  dt_gemm_wmma<<<gblk(64, 16), blk, 0, stream>>>(x_norm, Wq, nullptr, nullptr, qv, BX, DD, DD, 0);
  dt_l2norm<<<dim3(BX / 8), blk, 0, stream>>>(qv, BX);
  dt_state_read<<<dim3(BX), blk, tile_sh, stream>>>(state, qv, readv);
  dt_gemm_wmma<<<gblk(64, 16), blk, 0, stream>>>(readv, Wout, bout, u, xbuf, BX, DD, DD, 0);
  dt_layernorm<<<dim3(BX / 8), blk, 0, stream>>>(xbuf, n2g, n2b, x2, BX, 0);
  dt_gemm_wmma<<<gblk(64, 128), blk, 0, stream>>>(x2, W1, b1, nullptr, h1, BX, FFD, DD, 1);
  dt_gemm_wmma<<<gblk(64, 16),  blk, 0, stream>>>(h1, W2, b2, xbuf, out_emb, BX, DD, FFD, 0);
  dt_concat<<<dim3((BX * 512) / 256), blk, 0, stream>>>(x_norm, readv, meta_in);
  dt_gemm_wmma<<<gblk(64, 16), blk, 0, stream>>>(meta_in, Wm1, bm1, nullptr, hbuf, BX, DD, 2 * DD, 0);
  dt_layernorm<<<dim3(BX / 8), blk, 0, stream>>>(hbuf, mg, mb, hbuf, BX, 1);
  dt_gemm_wmma<<<gblk(64, 33), blk, 0, stream>>>(hbuf, Wm2, bm2, nullptr, meta_out, BX, N2, DD, 0);
  dt_mod<<<dim3((BX * DD) / 256), blk, 0, stream>>>(i_norm, meta_out, modv);
  dt_gemm_wmma<<<gblk(64, 16), blk, 0, stream>>>(modv, Wk, nullptr, nullptr, kv, BX, DD, DD, 0);
  dt_l2norm<<<dim3(BX / 8), blk, 0, stream>>>(kv, BX);
  dt_gemm_wmma<<<gblk(64, 16), blk, 0, stream>>>(modv, Wv, nullptr, nullptr, vv, BX, DD, DD, 0);
  dt_alpha_eta<<<dim3(BX / 8), blk, 0, stream>>>(modv, Wa, ba, We, be, meta_out, alpha, eta);
  dt_state_update<<<dim3(BX), blk, tile_sh, stream>>>(state, kv, vv, alpha, eta, out_state);
}